// EnvironmentLight_65953517797486
// MI455X (gfx1250) — compile-verified
//
#include <hip/hip_runtime.h>
#include <cstdint>

#define BLOCK 256

// ---------- small vector helpers ----------
__device__ __forceinline__ float3 ld3(const float* __restrict__ p) {
    return make_float3(p[0], p[1], p[2]);
}
__device__ __forceinline__ float dot3(float3 a, float3 b) {
    return a.x * b.x + a.y * b.y + a.z * b.z;
}

// ---------- cube face / uv (exact semantics of _cube_face_uv) ----------
__device__ __forceinline__ void cube_face_uv(float3 d, int& face, float& u, float& v) {
    float ax = fabsf(d.x), ay = fabsf(d.y), az = fabsf(d.z);
    bool is_x = (ax >= ay) && (ax >= az);
    bool is_y = (!is_x) && (ay >= az);
    float ma;
    if (is_x) {
        face = (d.x > 0.0f) ? 0 : 1;
        ma = ax;
        u = (d.x > 0.0f) ? -d.z : d.z;
        v = -d.y;
    } else if (is_y) {
        face = (d.y > 0.0f) ? 2 : 3;
        ma = ay;
        u = d.x;
        v = (d.y > 0.0f) ? d.z : -d.z;
    } else {
        face = (d.z > 0.0f) ? 4 : 5;
        ma = az;
        u = (d.z > 0.0f) ? d.x : -d.x;
        v = -d.y;
    }
    ma = fmaxf(ma, 1e-20f);
    u /= ma;
    v /= ma;
}

// ---------- bilinear gather on one cube face, 3 channels ----------
// tex layout: [6][W][W][3], row-major.  Matches _bilinear_gather clamp rules.
__device__ __forceinline__ float3 bilerp_face3(const float* __restrict__ tex, int W,
                                               int face, float u, float v) {
    float fx = (u * 0.5f + 0.5f) * (float)W - 0.5f;
    float fy = (v * 0.5f + 0.5f) * (float)W - 0.5f;
    float x0f = floorf(fx), y0f = floorf(fy);
    float tx = fx - x0f, ty = fy - y0f;
    int x0 = min(max((int)x0f, 0), W - 1);
    int x1 = min(x0 + 1, W - 1);
    int y0 = min(max((int)y0f, 0), W - 1);
    int y1 = min(y0 + 1, W - 1);
    int fb = face * W * W;
    const float* r0 = tex + (size_t)(fb + y0 * W) * 3;
    const float* r1 = tex + (size_t)(fb + y1 * W) * 3;
    float3 c00 = ld3(r0 + x0 * 3), c01 = ld3(r0 + x1 * 3);
    float3 c10 = ld3(r1 + x0 * 3), c11 = ld3(r1 + x1 * 3);
    float3 top = make_float3(c00.x * (1.0f - tx) + c01.x * tx,
                             c00.y * (1.0f - tx) + c01.y * tx,
                             c00.z * (1.0f - tx) + c01.z * tx);
    float3 bot = make_float3(c10.x * (1.0f - tx) + c11.x * tx,
                             c10.y * (1.0f - tx) + c11.y * tx,
                             c10.z * (1.0f - tx) + c11.z * tx);
    return make_float3(top.x * (1.0f - ty) + bot.x * ty,
                       top.y * (1.0f - ty) + bot.y * ty,
                       top.z * (1.0f - ty) + bot.z * ty);
}

// ---------- 256x256x2 FG LUT bilinear ----------
__device__ __forceinline__ float2 sample_lut2(const float* __restrict__ lut, float u, float v) {
    const int W = 256;
    float fx = u * (float)W - 0.5f;
    float fy = v * (float)W - 0.5f;
    float x0f = floorf(fx), y0f = floorf(fy);
    float tx = fx - x0f, ty = fy - y0f;
    int x0 = min(max((int)x0f, 0), W - 1);
    int x1 = min(x0 + 1, W - 1);
    int y0 = min(max((int)y0f, 0), W - 1);
    int y1 = min(y0 + 1, W - 1);
    const float* r0 = lut + (size_t)(y0 * W) * 2;
    const float* r1 = lut + (size_t)(y1 * W) * 2;
    float c00x = r0[x0 * 2 + 0], c00y = r0[x0 * 2 + 1];
    float c01x = r0[x1 * 2 + 0], c01y = r0[x1 * 2 + 1];
    float c10x = r1[x0 * 2 + 0], c10y = r1[x0 * 2 + 1];
    float c11x = r1[x1 * 2 + 0], c11y = r1[x1 * 2 + 1];
    float tpx = c00x * (1.0f - tx) + c01x * tx;
    float tpy = c00y * (1.0f - tx) + c01y * tx;
    float btx = c10x * (1.0f - tx) + c11x * tx;
    float bty = c10y * (1.0f - tx) + c11y * tx;
    float2 r;
    r.x = tpx * (1.0f - ty) + btx * ty;
    r.y = tpy * (1.0f - ty) + bty * ty;
    return r;
}

__device__ __forceinline__ const float* mip_sel(int l,
        const float* s0, const float* s1, const float* s2,
        const float* s3, const float* s4, const float* s5) {
    switch (l) {
        case 0: return s0;
        case 1: return s1;
        case 2: return s2;
        case 3: return s3;
        case 4: return s4;
        default: return s5;
    }
}

__device__ __forceinline__ float srgb1(float x) {
    return (x <= 0.0031308f) ? (12.92f * x)
                             : (1.055f * powf(fmaxf(x, 0.0031308f), 1.0f / 2.4f) - 0.055f);
}

// 6*16*16*3 floats = 18432 bytes = 1152 x b128 chunks
#define DM_FLOATS (6 * 16 * 16 * 3)
#define DM_CHUNKS ((DM_FLOATS * 4) / 16)

extern "C" __global__ void __launch_bounds__(BLOCK)
env_light_kernel(const float* __restrict__ view_dir, const float* __restrict__ normal,
                 const float* __restrict__ kd, const float* __restrict__ ks,
                 const float* __restrict__ occ, const float* __restrict__ diffuse_map,
                 const float* __restrict__ sp0, const float* __restrict__ sp1,
                 const float* __restrict__ sp2, const float* __restrict__ sp3,
                 const float* __restrict__ sp4, const float* __restrict__ sp5,
                 const float* __restrict__ fg_lut, float* __restrict__ out, int n) {
    // ---- async-stage the 18KB diffuse cubemap into LDS (CDNA5 async path) ----
    __shared__ float sdm[DM_FLOATS];
    {
        // Low 32 bits of a generic pointer into LDS == LDS byte address (ISA 10.2).
        uint32_t lds_base = (uint32_t)(uintptr_t)(void*)&sdm[0];
#pragma unroll
        for (int r = 0; r < 5; ++r) {
            int idx = (int)threadIdx.x + r * BLOCK;
            if (idx < DM_CHUNKS) {
                uint32_t boff = (uint32_t)idx * 16u;
                // GVS mode: mem = SADDR64 + VADDR32; dest LDS addr in first VGPR.
                asm volatile("global_load_async_to_lds_b128 %0, %1, %2"
                             :
                             : "v"(lds_base + boff), "v"(boff), "s"(diffuse_map)
                             : "memory");
            }
        }
        asm volatile("s_wait_asynccnt 0" ::: "memory");
        __syncthreads();
    }

    int i = (int)blockIdx.x * BLOCK + (int)threadIdx.x;
    if (i >= n) return;

    float3 vdir = ld3(view_dir + (size_t)i * 3);
    float3 nrm  = ld3(normal   + (size_t)i * 3);
    float3 kdv  = ld3(kd       + (size_t)i * 3);
    float3 ksv  = ld3(ks       + (size_t)i * 3);
    float  ro   = occ[i];

    float roughness = ksv.y;
    float metallic  = ksv.z;
    float3 spec_col = make_float3((1.0f - metallic) * 0.04f + kdv.x * metallic,
                                  (1.0f - metallic) * 0.04f + kdv.y * metallic,
                                  (1.0f - metallic) * 0.04f + kdv.z * metallic);
    float3 diff_col = make_float3(kdv.x * (1.0f - metallic),
                                  kdv.y * (1.0f - metallic),
                                  kdv.z * (1.0f - metallic));

    // reflect + safe normalize
    float dvn = dot3(vdir, nrm);
    float3 refl = make_float3(2.0f * dvn * nrm.x - vdir.x,
                              2.0f * dvn * nrm.y - vdir.y,
                              2.0f * dvn * nrm.z - vdir.z);
    float rlen2 = fmaxf(dot3(refl, refl), 1e-20f);
    float rinv = 1.0f / sqrtf(rlen2);
    refl = make_float3(refl.x * rinv, refl.y * rinv, refl.z * rinv);

    // ---- diffuse term: bilinear cubemap sample from LDS ----
    int dface; float du, dv;
    cube_face_uv(nrm, dface, du, dv);
    float3 diffuse = bilerp_face3(sdm, 16, dface, du, dv);
    diffuse.x = fmaxf(diffuse.x, 0.0f);
    diffuse.y = fmaxf(diffuse.y, 0.0f);
    diffuse.z = fmaxf(diffuse.z, 0.0f);

    float occl = 1.0f - ksv.x;
    float3 shaded = make_float3(diffuse.x * diff_col.x * occl,
                                diffuse.y * diff_col.y * occl,
                                diffuse.z * diff_col.z * occl);

    // ---- FG LUT ----
    float NdotV = fmaxf(dvn, 0.0001f);
    float2 fg = sample_lut2(fg_lut, NdotV, roughness);

    // ---- mip level (exact _get_mip) ----
    float lo = (fminf(fmaxf(roughness, 0.08f), 0.5f) - 0.08f) / (0.5f - 0.08f) * 4.0f;
    float hi = (fminf(fmaxf(roughness, 0.5f), 1.0f) - 0.5f) / 0.5f + 4.0f;
    float lvl = (roughness < 0.5f) ? lo : hi;
    lvl = fminf(fmaxf(lvl, 0.0f), 5.0f);
    int l0 = (int)floorf(lvl);
    int l1 = min(l0 + 1, 5);
    float lf = lvl - (float)l0;

    // ---- spec: trilinear cubemap across two mips ----
    int sface; float su, sv;
    cube_face_uv(refl, sface, su, sv);
    const float* t0 = mip_sel(l0, sp0, sp1, sp2, sp3, sp4, sp5);
    const float* t1 = mip_sel(l1, sp0, sp1, sp2, sp3, sp4, sp5);
    float3 sA = bilerp_face3(t0, 512 >> l0, sface, su, sv);
    float3 sB = bilerp_face3(t1, 512 >> l1, sface, su, sv);
    float3 spec = make_float3(sA.x * (1.0f - lf) + sB.x * lf,
                              sA.y * (1.0f - lf) + sB.y * lf,
                              sA.z * (1.0f - lf) + sB.z * lf);
    spec.x = fmaxf(spec.x, 0.0f);
    spec.y = fmaxf(spec.y, 0.0f);
    spec.z = fmaxf(spec.z, 0.0f);

    float3 reflectance = make_float3(spec_col.x * fg.x + fg.y,
                                     spec_col.y * fg.x + fg.y,
                                     spec_col.z * fg.x + fg.y);
    float ow = 1.0f - ro;
    shaded.x += spec.x * reflectance.x * ow;
    shaded.y += spec.y * reflectance.y * ow;
    shaded.z += spec.z * reflectance.z * ow;

    shaded.x = fminf(fmaxf(shaded.x, 0.0f), 1.0f);
    shaded.y = fminf(fmaxf(shaded.y, 0.0f), 1.0f);
    shaded.z = fminf(fmaxf(shaded.z, 0.0f), 1.0f);

    float* o = out + (size_t)i * 3;
    o[0] = srgb1(shaded.x);
    o[1] = srgb1(shaded.y);
    o[2] = srgb1(shaded.z);
}

extern "C" void kernel_launch(void* const* d_in, const int* in_sizes, int n_in,
                              void* d_out, int out_size, void* d_ws, size_t ws_size,
                              hipStream_t stream) {
    (void)n_in; (void)d_ws; (void)ws_size; (void)out_size;
    const float* view_dir    = (const float*)d_in[0];
    const float* normal      = (const float*)d_in[1];
    const float* kd          = (const float*)d_in[2];
    const float* ks          = (const float*)d_in[3];
    const float* occ         = (const float*)d_in[4];
    const float* diffuse_map = (const float*)d_in[5];
    const float* sp0         = (const float*)d_in[6];
    const float* sp1         = (const float*)d_in[7];
    const float* sp2         = (const float*)d_in[8];
    const float* sp3         = (const float*)d_in[9];
    const float* sp4         = (const float*)d_in[10];
    const float* sp5         = (const float*)d_in[11];
    const float* fg_lut      = (const float*)d_in[12];

    int n = in_sizes[0] / 3;
    dim3 grid((unsigned)((n + BLOCK - 1) / BLOCK)), block(BLOCK);
    hipLaunchKernelGGL(env_light_kernel, grid, block, 0, stream,
                       view_dir, normal, kd, ks, occ, diffuse_map,
                       sp0, sp1, sp2, sp3, sp4, sp5, fg_lut,
                       (float*)d_out, n);
}